// HyperDiffusion_56203942036041
// MI455X (gfx1250) — compile-verified
//
#include <hip/hip_runtime.h>

// HyperDiffusion on MI455X (gfx1250, wave32).
// d_in:  X [100000*128 f32], v_ids [1.6M i32], e_ids [1.6M i32], N_V (scalar), N_E (scalar)
// d_out: node_feat [100000*128 f32] ++ edge_feat [50000*128 f32]
// d_ws:  deg_v->inv_deg_v [100000 f32] ++ deg_e->inv_deg_e [50000 f32]

#define HG_NV 100000
#define HG_NE 50000
#define HG_C  128        // channels per row
#define HG_Q  (HG_C/4)   // float4 per row == 32 == wave32 lanes
#define HG_B  4          // incidence entries per wave (scatter) / per thread (degrees)

// ---------------------------------------------------------------------------
// Pass 1: degree counting. Each thread handles 4 consecutive entries via
// int4 (global_load_b128) index loads; degrees kept as float.
// ---------------------------------------------------------------------------
__global__ __launch_bounds__(256) void hg_degrees(
    const int* __restrict__ v_ids,
    const int* __restrict__ e_ids,
    float* __restrict__ deg_v,
    float* __restrict__ deg_e,
    int nnz) {
  int i0 = (blockIdx.x * blockDim.x + threadIdx.x) * HG_B;
  if (i0 >= nnz) return;
  if (i0 + HG_B <= nnz) {
    const int4 v4 = *(const int4*)(v_ids + i0);   // global_load_b128
    const int4 e4 = *(const int4*)(e_ids + i0);   // global_load_b128
    unsafeAtomicAdd(&deg_v[v4.x], 1.0f);
    unsafeAtomicAdd(&deg_v[v4.y], 1.0f);
    unsafeAtomicAdd(&deg_v[v4.z], 1.0f);
    unsafeAtomicAdd(&deg_v[v4.w], 1.0f);
    unsafeAtomicAdd(&deg_e[e4.x], 1.0f);
    unsafeAtomicAdd(&deg_e[e4.y], 1.0f);
    unsafeAtomicAdd(&deg_e[e4.z], 1.0f);
    unsafeAtomicAdd(&deg_e[e4.w], 1.0f);
  } else {
    for (int k = 0; k < HG_B && i0 + k < nnz; ++k) {
      unsafeAtomicAdd(&deg_v[v_ids[i0 + k]], 1.0f);
      unsafeAtomicAdd(&deg_e[e_ids[i0 + k]], 1.0f);
    }
  }
}

// ---------------------------------------------------------------------------
// Pass 1.5: deg -> 1/deg in place (150K elements).
// ---------------------------------------------------------------------------
__global__ __launch_bounds__(256) void hg_invert(float* __restrict__ deg, int n) {
  int i = blockIdx.x * blockDim.x + threadIdx.x;
  if (i >= n) return;
  float d = deg[i];
  deg[i] = (d > 0.0f) ? (1.0f / d) : 0.0f;
}

// ---------------------------------------------------------------------------
// Scatter body shared by v2e and e2v:
//   dst_feat[dst_ids[w]] += SRC4[src_ids[w]] * inv_deg[src_ids[w]]
// One wave32 per HG_B consecutive entries; lane L owns floats [4L,4L+3] of
// the 128-ch row. Entry indices are wave-uniform (readfirstlane) so the
// index/scale loads are scalar (contiguous -> mergeable into s_load_b128).
// Fast path (all waves except at most the last): no per-entry guards --
// straight-line load phase (4 independent global_load_b128 in flight) then
// atomic phase (16 global_atomic_add_f32), so gather latency of entry k+1
// overlaps atomic drain of entry k inside the wave.
// ---------------------------------------------------------------------------
__device__ __forceinline__ void hg_scatter_entry(
    const float4* __restrict__ SRC4, const float* __restrict__ inv_deg,
    float* __restrict__ dst_feat, int src, int dst, int lane,
    float4* xout, float* sout) {
  *sout = inv_deg[src];                          // uniform -> scalar load
  *xout = SRC4[(size_t)src * HG_Q + lane];       // global_load_b128
  (void)dst_feat; (void)dst;
}

__device__ __forceinline__ void hg_scatter_flush(
    float* __restrict__ dst_feat, int dst, int lane, float4 x, float s) {
  float* d = dst_feat + (size_t)dst * HG_C + (lane << 2);
  unsafeAtomicAdd(d + 0, x.x * s);
  unsafeAtomicAdd(d + 1, x.y * s);
  unsafeAtomicAdd(d + 2, x.z * s);
  unsafeAtomicAdd(d + 3, x.w * s);
}

__device__ __forceinline__ void hg_scatter_body(
    const float4* __restrict__ SRC4,      // [*, HG_Q] float4 rows
    const int*    __restrict__ src_ids,
    const int*    __restrict__ dst_ids,
    const float*  __restrict__ inv_deg,   // indexed by src id
    float*        __restrict__ dst_feat,
    int nnz) {
  const int lane = threadIdx.x & 31;
  const int wave = __builtin_amdgcn_readfirstlane(
      (int)(blockIdx.x * (blockDim.x >> 5) + (threadIdx.x >> 5)));
  const int wbase = wave * HG_B;
  if (wbase >= nnz) return;                      // uniform whole-wave exit

  int    src[HG_B], dst[HG_B];
  float  s[HG_B];
  float4 x[HG_B];

  if (wbase + HG_B <= nnz) {
    // ---- fast path: full batch, no per-entry guards ----
#pragma unroll
    for (int k = 0; k < HG_B; ++k) {
      src[k] = src_ids[wbase + k];               // scalar, contiguous
      dst[k] = dst_ids[wbase + k];               // scalar, contiguous
    }
#pragma unroll
    for (int k = 0; k < HG_B; ++k)
      hg_scatter_entry(SRC4, inv_deg, dst_feat, src[k], dst[k], lane,
                       &x[k], &s[k]);
#pragma unroll
    for (int k = 0; k < HG_B; ++k)
      hg_scatter_flush(dst_feat, dst[k], lane, x[k], s[k]);
  } else {
    // ---- tail path: at most one wave in the whole grid ----
    for (int k = 0; k < HG_B && wbase + k < nnz; ++k) {
      const int sv = src_ids[wbase + k];
      const int dv = dst_ids[wbase + k];
      float  sk; float4 xk;
      hg_scatter_entry(SRC4, inv_deg, dst_feat, sv, dv, lane, &xk, &sk);
      hg_scatter_flush(dst_feat, dv, lane, xk, sk);
    }
  }
}

// Pass 2 (v2e): edge_feat[e] += X[v] * inv_deg_v[v]
__global__ __launch_bounds__(256) void hg_v2e(
    const float4* __restrict__ X4,
    const int*    __restrict__ v_ids,
    const int*    __restrict__ e_ids,
    const float*  __restrict__ inv_deg_v,
    float*        __restrict__ edge_feat,
    int nnz) {
  hg_scatter_body(X4, v_ids, e_ids, inv_deg_v, edge_feat, nnz);
}

// Pass 3 (e2v): node_feat[v] += edge_feat[e] * inv_deg_e[e]
__global__ __launch_bounds__(256) void hg_e2v(
    const float4* __restrict__ EF4,
    const int*    __restrict__ e_ids,
    const int*    __restrict__ v_ids,
    const float*  __restrict__ inv_deg_e,
    float*        __restrict__ node_feat,
    int nnz) {
  hg_scatter_body(EF4, e_ids, v_ids, inv_deg_e, node_feat, nnz);
}

// ---------------------------------------------------------------------------
extern "C" void kernel_launch(void* const* d_in, const int* in_sizes, int n_in,
                              void* d_out, int out_size, void* d_ws, size_t ws_size,
                              hipStream_t stream) {
  const float* X     = (const float*)d_in[0];
  const int*   v_ids = (const int*)d_in[1];
  const int*   e_ids = (const int*)d_in[2];
  const int    nnz   = in_sizes[1];

  float* node_feat = (float*)d_out;                       // NV * C
  float* edge_feat = node_feat + (size_t)HG_NV * HG_C;    // NE * C
  float* deg_v     = (float*)d_ws;                        // NV (becomes inv_deg_v)
  float* deg_e     = deg_v + HG_NV;                       // NE (becomes inv_deg_e)

  // Zero accumulators (harness poisons d_out/d_ws; identical work every call).
  hipMemsetAsync(d_out, 0, (size_t)(HG_NV + HG_NE) * HG_C * sizeof(float), stream);
  hipMemsetAsync(d_ws,  0, (size_t)(HG_NV + HG_NE) * sizeof(float), stream);

  // Pass 1: degrees (4 entries per thread).
  {
    int threads = (nnz + HG_B - 1) / HG_B;
    hg_degrees<<<(threads + 255) / 256, 256, 0, stream>>>(v_ids, e_ids,
                                                          deg_v, deg_e, nnz);
  }

  // Pass 1.5: invert both degree arrays in place (contiguous in ws).
  {
    int n = HG_NV + HG_NE;
    hg_invert<<<(n + 255) / 256, 256, 0, stream>>>(deg_v, n);
  }

  // Passes 2 & 3: one wave32 per 4 entries, 8 waves per 256-thread block
  // -> 32 entries per block.
  {
    int blocks = (nnz + 8 * HG_B - 1) / (8 * HG_B);
    hg_v2e<<<blocks, 256, 0, stream>>>((const float4*)X, v_ids, e_ids,
                                       deg_v, edge_feat, nnz);
    hg_e2v<<<blocks, 256, 0, stream>>>((const float4*)edge_feat, e_ids, v_ids,
                                       deg_e, node_feat, nnz);
  }
}